// PointWarping_9354438770941
// MI455X (gfx1250) — compile-verified
//
#include <hip/hip_runtime.h>

typedef __attribute__((ext_vector_type(2))) float v2f;
typedef __attribute__((ext_vector_type(8))) float v8f;

#define EPSD   1e-10f
#define CLAMPV 10.0f

// ---------------------------------------------------------------------------
// Phase 1: build WMMA-friendly database + flow-gather arrays.
//   db[b*N+n]  = (-2kx, -2ky, -2kz, |k|^2)  with k = pos1 + flow1  (A operand)
//   flw[b*N+n] = (fx, fy, fz, 0)                                  (gather src)
// NOTE: flw is laid out immediately after db in the workspace; the NN kernel
// relies on this to over-read one tile past db's end (software pipeline).
// ---------------------------------------------------------------------------
__global__ __launch_bounds__(256) void pw_prep(const float* __restrict__ pos1,
                                               const float* __restrict__ flow1,
                                               float4* __restrict__ db,
                                               float4* __restrict__ flw,
                                               int B, int N) {
  int i = blockIdx.x * blockDim.x + threadIdx.x;
  if (i >= B * N) return;
  int b = i / N, n = i - b * N;
  const float* p = pos1 + (size_t)b * 3 * N;
  const float* f = flow1 + (size_t)b * 3 * N;
  float fx = f[n], fy = f[n + N], fz = f[n + 2 * N];
  float kx = p[n] + fx, ky = p[n + N] + fy, kz = p[n + 2 * N] + fz;
  db[i]  = make_float4(-2.f * kx, -2.f * ky, -2.f * kz, kx * kx + ky * ky + kz * kz);
  flw[i] = make_float4(fx, fy, fz, 0.f);
}

// Branch-free sorted insert into a per-lane top-3 (ascending) list.
__device__ __forceinline__ void top3_insert(float s, int m,
                                            float& d0, float& d1, float& d2,
                                            int& i0, int& i1, int& i2) {
  bool lt2 = s < d2, lt1 = s < d1, lt0 = s < d0;
  float nd2 = lt1 ? d1 : (lt2 ? s : d2); int ni2 = lt1 ? i1 : (lt2 ? m : i2);
  float nd1 = lt0 ? d0 : (lt1 ? s : d1); int ni1 = lt0 ? i0 : (lt1 ? m : i1);
  float nd0 = lt0 ? s  : d0;             int ni0 = lt0 ? m  : i0;
  d0 = nd0; d1 = nd1; d2 = nd2; i0 = ni0; i1 = ni1; i2 = ni2;
}

// ---------------------------------------------------------------------------
// Phase 2: per wave, 16 queries; sweep all N1 database points in 16-wide tiles
// scored by v_wmma_f32_16x16x4_f32 (score = -2 q.k + |k|^2, rank-equivalent to
// d^2 per query); per-lane top-3; shfl-merge halves; exact-distance IDW.
//
// WMMA operand layout (wave32, f32 K=4):
//   A (16x4, DB tile):  lane<16 : VGPR0=K0, VGPR1=K1 for M=lane
//                       lane>=16: VGPR0=K2, VGPR1=K3 for M=lane-16
//     -> float4 db point loads directly as v2f at offset 0 (lo) / 8B (hi).
//   B (4x16, queries):  lane<16 : VGPR0=K0(=qx), VGPR1=K1(=qy), N=lane
//                       lane>=16: VGPR0=K2(=qz), VGPR1=K3(=1.0), N=lane-16
//   D (16x16):          lane holds 8 scores of query N=(lane&15), rows
//                       M = vgpr + (lane>=16 ? 8 : 0).
// ---------------------------------------------------------------------------
__global__ __launch_bounds__(256) void pw_nn3(const float* __restrict__ pos2,
                                              const float4* __restrict__ db,
                                              const float4* __restrict__ flw,
                                              float* __restrict__ out,
                                              int B, int N1, int N2) {
  const int lane  = threadIdx.x & 31;
  const int wave  = threadIdx.x >> 5;
  const int wid   = blockIdx.x * (blockDim.x >> 5) + wave;
  const int tilesQ = N2 >> 4;
  const int b  = wid / tilesQ;          // uniform per wave
  const int qt = wid - b * tilesQ;
  if (b >= B) return;                   // uniform; never taken with exact grid

  const int q = qt * 16 + (lane & 15);
  const float* p2 = pos2 + (size_t)b * 3 * N2;
  const float qx = p2[q], qy = p2[q + N2], qz = p2[q + 2 * N2];

  const bool hi = lane >= 16;
  v2f bq;                                // query operand (loop invariant)
  bq.x = hi ? qz : qx;
  bq.y = hi ? 1.0f : qy;

  const float4* dbB = db + (size_t)b * N1;
  const float*  dbF = (const float*)dbB;
  const int fOff  = (lane & 15) * 4 + (hi ? 2 : 0);  // float offset of A fragment
  const int mhalf = hi ? 8 : 0;

  float d0 = 3.4e38f, d1 = 3.4e38f, d2 = 3.4e38f;
  int   i0 = 0, i1 = 0, i2 = 0;

  const int nt = N1 >> 4;

  // Software pipeline: A fragment for tile t+1 is loaded before tile t's
  // selection work, so the s_wait for it sits behind ~120 VALU ops.
  // The t==nt-1 over-read lands in flw (contiguous in the workspace): safe.
  v2f a = *(const v2f*)(dbF + fOff);
  for (int t = 0; t < nt; ++t) {
    v2f an = *(const v2f*)(dbF + (size_t)(t + 1) * 64 + fOff);
    __builtin_prefetch(dbF + (size_t)(t + 16) * 64, 0, 3);  // WGP-scope, 4KB ahead
    v8f c = {};
    c = __builtin_amdgcn_wmma_f32_16x16x4_f32(
        /*neg_a=*/false, a, /*neg_b=*/false, bq,
        /*c_mod=*/(short)0, c, /*reuse_a=*/false, /*reuse_b=*/false);
    const int mbase = t * 16 + mhalf;
#pragma unroll
    for (int v = 0; v < 8; ++v) {
      top3_insert(c[v], mbase + v, d0, d1, d2, i0, i1, i2);
    }
    a = an;
  }

  // Merge the two half-lane partial top-3 lists (partner = lane ^ 16).
  {
    float pd0 = __shfl_xor(d0, 16, 32), pd1 = __shfl_xor(d1, 16, 32),
          pd2 = __shfl_xor(d2, 16, 32);
    int   pi0 = __shfl_xor(i0, 16, 32), pi1 = __shfl_xor(i1, 16, 32),
          pi2 = __shfl_xor(i2, 16, 32);
    top3_insert(pd0, pi0, d0, d1, d2, i0, i1, i2);
    top3_insert(pd1, pi1, d0, d1, d2, i0, i1, i2);
    top3_insert(pd2, pi2, d0, d1, d2, i0, i1, i2);
  }

  if (lane < 16) {
    const int idx[3] = {i0, i1, i2};
    float wts[3], wsum = 0.f;
#pragma unroll
    for (int j = 0; j < 3; ++j) {
      float4 dv = dbB[idx[j]];
      float kx = -0.5f * dv.x, ky = -0.5f * dv.y, kz = -0.5f * dv.z;
      float dx = kx - qx, dy = ky - qy, dz = kz - qz;
      float dist = sqrtf(dx * dx + dy * dy + dz * dz);
      dist = fmaxf(dist, EPSD);
      wts[j] = 1.0f / dist;
      wsum += wts[j];
    }
    float rws = 1.0f / wsum;
    float fxs = 0.f, fys = 0.f, fzs = 0.f;
    const float4* flB = flw + (size_t)b * N1;
#pragma unroll
    for (int j = 0; j < 3; ++j) {
      float4 fv = flB[idx[j]];
      float w = wts[j] * rws;
      fxs += w * fv.x; fys += w * fv.y; fzs += w * fv.z;
    }
    float* o = out + (size_t)b * 3 * N2;
    o[q]          = fminf(fmaxf(qx - fxs, -CLAMPV), CLAMPV);
    o[q + N2]     = fminf(fmaxf(qy - fys, -CLAMPV), CLAMPV);
    o[q + 2 * N2] = fminf(fmaxf(qz - fzs, -CLAMPV), CLAMPV);
  }
}

extern "C" void kernel_launch(void* const* d_in, const int* in_sizes, int n_in,
                              void* d_out, int out_size, void* d_ws, size_t ws_size,
                              hipStream_t stream) {
  const float* pos1  = (const float*)d_in[0];
  const float* pos2  = (const float*)d_in[1];
  const float* flow1 = (const float*)d_in[2];

  const int B = 4, C = 3;
  const int N = in_sizes[0] / (B * C);          // 8192

  float4* db  = (float4*)d_ws;                  // B*N*16 bytes
  float4* flw = db + (size_t)B * N;             // B*N*16 bytes (directly after db)

  int total = B * N;
  pw_prep<<<(total + 255) / 256, 256, 0, stream>>>(pos1, flow1, db, flw, B, N);

  int waves  = B * (N / 16);                    // 2048 waves, 16 queries each
  int blocks = waves / 8;                       // 256 threads = 8 waves/block
  pw_nn3<<<blocks, 256, 0, stream>>>(pos2, db, flw, (float*)d_out, B, N, N);
}